// SGC_LSTM_83056077570605
// MI455X (gfx1250) — compile-verified
//
#include <hip/hip_runtime.h>
#include <math.h>

// ---------------------------------------------------------------------------
// SGC+LSTM for MI455X (gfx1250, wave32).
// Phase 1: tiny-graph signed SAGE -> VALU/LDS, 1 WG per (g,m,v) context.
// Phase 2: 25-step LSTM, gates GEMM [20480 x 194] x [194 x 512] per step via
//          V_WMMA_F32_16X16X4_F32 (fp32 tensor path, matches fp32 reference).
//          32-row M-tiles per WG (2x B-fragment reuse), fragment-ready packed
//          weights (one global b64 load per B fragment), fused in-register
//          gate pointwise, in-place hx/cx (row-disjoint WGs).
// ---------------------------------------------------------------------------

#define G_     2048
#define M_     5
#define V_     5
#define P_     10
#define E2_    32
#define D_     32
#define H_     128
#define STEPS_ 25
#define FEATP  68          // FEAT=66 padded to multiple of 4 (zero-padded)
#define BATCH  (G_ * P_)   // 20480
#define GATES  (4 * H_)    // 512
#define NGX    (FEATP / 4) // 17 K-groups for x part
#define NGH    (H_ / 4)    // 32 K-groups for h part
#define HPAD   132         // LDS row stride for hx tile (bank rotation)

typedef __attribute__((ext_vector_type(2))) float v2f;
typedef __attribute__((ext_vector_type(8))) float v8f;

static __device__ __forceinline__ v8f wmma4(v2f a, v2f b, v8f c) {
  // D = A(16x4 f32) * B(4x16 f32) + C(16x16 f32)
  return __builtin_amdgcn_wmma_f32_16x16x4_f32(
      /*neg_a=*/false, a, /*neg_b=*/false, b,
      /*c_mod=*/(short)0, c, /*reuse_a=*/false, /*reuse_b=*/false);
}

static __device__ __forceinline__ float sigmoidf_(float x) {
  return 1.0f / (1.0f + __expf(-x));
}

// ---------------------------------------------------------------------------
// Weight prep into fragment-ready packed layouts:
//   wbih_p[t][gi<17][half<2][n<512][2] : pair (B[ka][n], B[ka+1][n]) where
//     B[k][n] = w_ih[t][n][k] (zero for k >= 66), ka = 4*gi + 2*half.
//   wbhh_p[t][gi<32][half<2][n<512][2] : same for w_hh (K = 128).
//   bsum = b_ih + b_hh.
// One lane's B fragment = a single 8-byte load; half-waves read dense 128B.
// ---------------------------------------------------------------------------
__global__ void prep_kernel(const float* __restrict__ w_ih,
                            const float* __restrict__ w_hh,
                            const float* __restrict__ b_ih,
                            const float* __restrict__ b_hh,
                            float* __restrict__ wbih_p,
                            float* __restrict__ wbhh_p,
                            float* __restrict__ bsum) {
  const int stride = gridDim.x * blockDim.x;
  const int tid0 = blockIdx.x * blockDim.x + threadIdx.x;

  const int tot1 = STEPS_ * NGX * 2 * GATES * 2;     // 870400
  for (int i = tid0; i < tot1; i += stride) {
    const int j = i & 1;
    const int n = (i >> 1) & 511;
    const int half = (i >> 10) & 1;
    const int rest = i >> 11;          // t*17 + gi
    const int gi = rest % NGX;
    const int t = rest / NGX;
    const int k = 4 * gi + 2 * half + j;
    wbih_p[i] = (k < 66) ? w_ih[((size_t)t * GATES + n) * 66 + k] : 0.0f;
  }
  const int tot2 = STEPS_ * NGH * 2 * GATES * 2;     // 1638400
  for (int i = tid0; i < tot2; i += stride) {
    const int j = i & 1;
    const int n = (i >> 1) & 511;
    const int half = (i >> 10) & 1;
    const int rest = i >> 11;          // t*32 + gi
    const int gi = rest & 31;
    const int t = rest >> 5;
    const int k = 4 * gi + 2 * half + j;
    wbhh_p[i] = w_hh[((size_t)t * GATES + n) * H_ + k];
  }
  const int tot3 = STEPS_ * GATES;
  for (int i = tid0; i < tot3; i += stride) bsum[i] = b_ih[i] + b_hh[i];
}

// ---------------------------------------------------------------------------
// GNN: one workgroup per (g, m, v). 320 threads = 10 waves; wave p owns graph
// row p, lane d owns feature channel d. Writes x_ws[mv][g*P+p][0..67].
// ---------------------------------------------------------------------------
__global__ __launch_bounds__(320) void gnn_kernel(
    const float* __restrict__ h0, const float* __restrict__ add_info,
    const float* __restrict__ wpb, const float* __restrict__ bpb,
    const float* __restrict__ wnb, const float* __restrict__ bnb,
    const float* __restrict__ wpd, const float* __restrict__ bpd,
    const float* __restrict__ wnd, const float* __restrict__ bnd,
    const int* __restrict__ pos_adj, const int* __restrict__ neg_adj,
    float* __restrict__ x_ws) {
  __shared__ float sAp[P_][P_], sAn[P_][P_];
  __shared__ float sdeg[4][P_];          // rowP, rowN, colP, colN
  __shared__ float sh0[P_][E2_];
  __shared__ float shp[P_][E2_], shn[P_][E2_];
  __shared__ float scat[P_][6 * D_];

  const int ctx = blockIdx.x;            // (g*M + m)*V + v
  const int t = threadIdx.x;
  const int p = t >> 5;
  const int d = t & 31;
  const int g = ctx / (M_ * V_);
  const int mv = ctx % (M_ * V_);
  const int m = mv / V_;

  const size_t hb = (size_t)ctx * P_ * E2_;
  const size_t ab = (size_t)ctx * P_ * P_;

  sh0[p][d] = h0[hb + p * E2_ + d];
  if (t < 100) {
    const int i = t / 10, j = t % 10;
    sAp[i][j] = (i != j) ? (float)pos_adj[ab + i * 10 + j] : 0.0f;
  }
  if (t >= 128 && t < 228) {
    const int tt = t - 128, i = tt / 10, j = tt % 10;
    sAn[i][j] = (i != j) ? (float)neg_adj[ab + i * 10 + j] : 0.0f;
  }
  __syncthreads();
  if (t < P_)                       { float s = 0; for (int j = 0; j < P_; ++j) s += sAp[t][j];        sdeg[0][t] = s; }
  else if (t >= 32 && t < 32 + P_)  { const int i = t - 32; float s = 0; for (int j = 0; j < P_; ++j) s += sAn[i][j]; sdeg[1][i] = s; }
  else if (t >= 64 && t < 64 + P_)  { const int i = t - 64; float s = 0; for (int j = 0; j < P_; ++j) s += sAp[j][i]; sdeg[2][i] = s; }
  else if (t >= 96 && t < 96 + P_)  { const int i = t - 96; float s = 0; for (int j = 0; j < P_; ++j) s += sAn[j][i]; sdeg[3][i] = s; }
  __syncthreads();

  // ---- base layer ----
  {
    float ap = 0, an = 0;
    for (int j = 0; j < P_; ++j) {
      const float hj = sh0[j][d];
      ap += sAp[p][j] * hj;
      an += sAn[p][j] * hj;
    }
    scat[p][d]      = ap / fmaxf(sdeg[0][p], 1.0f);
    scat[p][32 + d] = an / fmaxf(sdeg[1][p], 1.0f);
  }
  __syncthreads();
  {
    float accp = bpb[d], accn = bnb[d];
    for (int k = 0; k < E2_; ++k) {
      const float mp = scat[p][k], mn = scat[p][32 + k], hs = sh0[p][k];
      accp += mp * wpb[k * D_ + d] + hs * wpb[(E2_ + k) * D_ + d];
      accn += mn * wnb[k * D_ + d] + hs * wnb[(E2_ + k) * D_ + d];
    }
    float sp = accp * accp, sn = accn * accn;
    for (int mm = 16; mm >= 1; mm >>= 1) {
      sp += __shfl_xor(sp, mm, 32);
      sn += __shfl_xor(sn, mm, 32);
    }
    shp[p][d] = accp / fmaxf(sqrtf(sp), 1e-12f);
    shn[p][d] = accn / fmaxf(sqrtf(sn), 1e-12f);
  }
  __syncthreads();

  // ---- two deep layers ----
  for (int layer = 0; layer < 2; ++layer) {
    float b0 = 0, b1 = 0, b2 = 0, b3 = 0, b4 = 0, b5 = 0;
    for (int j = 0; j < P_; ++j) {
      const float ap = sAp[p][j], an = sAn[p][j];
      const float apT = sAp[j][p], anT = sAn[j][p];
      const float hpj = shp[j][d], hnj = shn[j][d];
      b0 += ap * hpj;  b1 += an * hnj;  b2 += ap * hnj;
      b3 += an * hpj;  b4 += apT * hpj; b5 += anT * hnj;
    }
    b0 /= fmaxf(sdeg[0][p], 1.0f); b1 /= fmaxf(sdeg[1][p], 1.0f);
    b2 /= fmaxf(sdeg[0][p], 1.0f); b3 /= fmaxf(sdeg[1][p], 1.0f);
    b4 /= fmaxf(sdeg[2][p], 1.0f); b5 /= fmaxf(sdeg[3][p], 1.0f);
    scat[p][0 * D_ + d] = b0; scat[p][1 * D_ + d] = b1; scat[p][2 * D_ + d] = b2;
    scat[p][3 * D_ + d] = b3; scat[p][4 * D_ + d] = b4; scat[p][5 * D_ + d] = b5;
    __syncthreads();
    float accp = bpd[layer * D_ + d], accn = bnd[layer * D_ + d];
    const float* wp = wpd + (size_t)layer * 224 * D_;
    const float* wn = wnd + (size_t)layer * 224 * D_;
    for (int k = 0; k < 6 * D_; ++k) {
      const float c = scat[p][k];
      accp += c * wp[k * D_ + d];
      accn += c * wn[k * D_ + d];
    }
    for (int k = 0; k < D_; ++k) {
      accp += shp[p][k] * wp[(6 * D_ + k) * D_ + d];
      accn += shn[p][k] * wn[(6 * D_ + k) * D_ + d];
    }
    __syncthreads();  // all reads of shp/shn done before overwrite
    float sp = accp * accp, sn = accn * accn;
    for (int mm = 16; mm >= 1; mm >>= 1) {
      sp += __shfl_xor(sp, mm, 32);
      sn += __shfl_xor(sn, mm, 32);
    }
    shp[p][d] = accp / fmaxf(sqrtf(sp), 1e-12f);
    shn[p][d] = accn / fmaxf(sqrtf(sn), 1e-12f);
    __syncthreads();
  }

  // ---- emit x[step = mv][row = g*P+p][0..67] ----
  {
    const size_t row = (size_t)g * P_ + p;
    float* xo = x_ws + ((size_t)mv * BATCH + row) * FEATP;
    xo[d]       = shp[p][d];
    xo[D_ + d]  = shn[p][d];
    if (d < 2) xo[64 + d] = add_info[(((size_t)g * M_ + m) * P_ + p) * 2 + d];
    if (d == 2 || d == 3) xo[64 + d] = 0.0f;  // K-padding
  }
}

// ---------------------------------------------------------------------------
// One LSTM step. Grid 640 (32 batch rows each), 256 threads (8 waves).
// Wave w owns H-columns [16w,16w+16); it computes 8 output tiles:
// 2 row-tiles x 4 gate tiles (i/f/g/o are 128 apart), so every B fragment
// (one b64 load) feeds 2 WMMAs and the whole gate pointwise stays in-register.
// ---------------------------------------------------------------------------
__global__ __launch_bounds__(256) void lstm_step_kernel(
    const float* __restrict__ x_ws,     // [25][20480][68]
    const float* __restrict__ wbih_p,   // [25][17][2][512][2]
    const float* __restrict__ wbhh_p,   // [25][32][2][512][2]
    const float* __restrict__ bsum,     // [25][512]
    float* __restrict__ hx, float* __restrict__ cx, int t) {
  __shared__ float shX[32 * FEATP];     // stride 68 -> rows rotate LDS banks
  __shared__ float shH[32 * HPAD];      // 128 padded to 132 (bank rotation)

  const int tid = threadIdx.x;
  const int r0 = blockIdx.x << 5;       // 32 rows per WG

  // stage X tile (contiguous in global -> vector copies)
  const float* xsrc = x_ws + ((size_t)t * BATCH + r0) * FEATP;
  for (int i = tid; i < 32 * FEATP / 2; i += 256)
    ((v2f*)shX)[i] = ((const v2f*)xsrc)[i];
  // stage H tile with padded row stride
  for (int i = tid; i < 32 * (H_ / 2); i += 256) {
    const int r = i >> 6;               // / 64 pairs per row
    const int c = (i & 63) << 1;
    *(v2f*)&shH[r * HPAD + c] = *(const v2f*)&hx[(size_t)(r0 + r) * H_ + c];
  }
  __syncthreads();

  const int lane = tid & 31;
  const int wave = tid >> 5;
  const int half = lane >> 4;           // 0: K rows k0,k0+1 ; 1: rows k0+2,+3
  const int l = lane & 15;
  const int cb = wave << 4;             // column base within H

  v8f acc[4][2];
  {
    const v8f z = {};
#pragma unroll
    for (int q = 0; q < 4; ++q) { acc[q][0] = z; acc[q][1] = z; }
  }

  // ---- x @ w_ih^T (K = 68, padded rows are zero) ----
  {
    const float* wxp = wbih_p + (size_t)t * (NGX * 2048) + half * 1024;
#pragma unroll 4
    for (int gi = 0; gi < NGX; ++gi) {
      const int ka = 4 * gi + (half << 1);
      const v2f a0 = *(const v2f*)&shX[l * FEATP + ka];
      const v2f a1 = *(const v2f*)&shX[(16 + l) * FEATP + ka];
      const float* wb = wxp + (size_t)gi * 2048;
#pragma unroll
      for (int q = 0; q < 4; ++q) {
        const v2f b = *(const v2f*)&wb[(q * 128 + cb + l) * 2];
        acc[q][0] = wmma4(a0, b, acc[q][0]);
        acc[q][1] = wmma4(a1, b, acc[q][1]);
      }
    }
  }
  // ---- hx @ w_hh^T (K = 128) ----
  {
    const float* whp = wbhh_p + (size_t)t * (NGH * 2048) + half * 1024;
#pragma unroll 4
    for (int gi = 0; gi < NGH; ++gi) {
      const int ka = 4 * gi + (half << 1);
      const v2f a0 = *(const v2f*)&shH[l * HPAD + ka];
      const v2f a1 = *(const v2f*)&shH[(16 + l) * HPAD + ka];
      const float* wb = whp + (size_t)gi * 2048;
#pragma unroll
      for (int q = 0; q < 4; ++q) {
        const v2f b = *(const v2f*)&wb[(q * 128 + cb + l) * 2];
        acc[q][0] = wmma4(a0, b, acc[q][0]);
        acc[q][1] = wmma4(a1, b, acc[q][1]);
      }
    }
  }

  // ---- fused bias + gates + state update (all gates in-register) ----
  const float* bs = bsum + (size_t)t * GATES;
  float bb[4];
#pragma unroll
  for (int q = 0; q < 4; ++q) bb[q] = bs[q * 128 + cb + l];
#pragma unroll
  for (int rt = 0; rt < 2; ++rt) {
#pragma unroll
    for (int r = 0; r < 8; ++r) {
      const int row = r0 + 16 * rt + r + (half << 3);  // C/D: VGPR r -> M=r/r+8
      const size_t idx = (size_t)row * H_ + cb + l;
      const float ig = sigmoidf_(acc[0][rt][r] + bb[0]);
      const float fg = sigmoidf_(acc[1][rt][r] + bb[1]);
      const float gg = tanhf(acc[2][rt][r] + bb[2]);
      const float og = sigmoidf_(acc[3][rt][r] + bb[3]);
      const float c = fg * cx[idx] + ig * gg;
      cx[idx] = c;
      hx[idx] = og * tanhf(c);
    }
  }
}

// ---------------------------------------------------------------------------
extern "C" void kernel_launch(void* const* d_in, const int* in_sizes, int n_in,
                              void* d_out, int out_size, void* d_ws,
                              size_t ws_size, hipStream_t stream) {
  (void)in_sizes; (void)n_in; (void)out_size; (void)ws_size;
  const float* h0        = (const float*)d_in[0];
  const float* add_info  = (const float*)d_in[1];
  const float* wpb       = (const float*)d_in[2];
  const float* bpb       = (const float*)d_in[3];
  const float* wnb       = (const float*)d_in[4];
  const float* bnb       = (const float*)d_in[5];
  const float* wpd       = (const float*)d_in[6];
  const float* bpd       = (const float*)d_in[7];
  const float* wnd       = (const float*)d_in[8];
  const float* bnd       = (const float*)d_in[9];
  const float* w_ih      = (const float*)d_in[10];
  const float* w_hh      = (const float*)d_in[11];
  const float* b_ih      = (const float*)d_in[12];
  const float* b_hh      = (const float*)d_in[13];
  const float* hx0       = (const float*)d_in[14];
  const float* cx0       = (const float*)d_in[15];
  const int*   pos_adj   = (const int*)d_in[16];
  const int*   neg_adj   = (const int*)d_in[17];

  float* ws = (float*)d_ws;
  size_t off = 0;
  float* x_ws   = ws + off; off += (size_t)STEPS_ * BATCH * FEATP;   // 34.8M f
  float* wbih_p = ws + off; off += (size_t)STEPS_ * NGX * 2048;
  float* wbhh_p = ws + off; off += (size_t)STEPS_ * NGH * 2048;
  float* bsum   = ws + off; off += (size_t)STEPS_ * GATES;
  float* hx     = ws + off; off += (size_t)BATCH * H_;
  float* cx     = ws + off; off += (size_t)BATCH * H_;

  hipMemcpyAsync(hx, hx0, sizeof(float) * BATCH * H_,
                 hipMemcpyDeviceToDevice, stream);
  hipMemcpyAsync(cx, cx0, sizeof(float) * BATCH * H_,
                 hipMemcpyDeviceToDevice, stream);

  prep_kernel<<<1024, 256, 0, stream>>>(w_ih, w_hh, b_ih, b_hh, wbih_p,
                                        wbhh_p, bsum);
  gnn_kernel<<<G_ * M_ * V_, 320, 0, stream>>>(
      h0, add_info, wpb, bpb, wnb, bnb, wpd, bpd, wnd, bnd, pos_adj, neg_adj,
      x_ws);
  for (int t = 0; t < STEPS_; ++t) {
    lstm_step_kernel<<<BATCH / 32, 256, 0, stream>>>(x_ws, wbih_p, wbhh_p,
                                                     bsum, hx, cx, t);
  }
  hipMemcpyAsync(d_out, hx, sizeof(float) * BATCH * H_,
                 hipMemcpyDeviceToDevice, stream);
}